// Social_Encoder_19112604467372
// MI455X (gfx1250) — compile-verified
//
#include <hip/hip_runtime.h>
#include <hip/hip_bf16.h>

typedef __attribute__((ext_vector_type(2))) float v2f;
typedef __attribute__((ext_vector_type(8))) float v8f;

#define EMBED      64
#define KDIM       128     // 2*EMBED
#define TILE_M     16
#define LDS_STRIDE 130     // 128 + 2 pad floats: breaks bank aliasing, keeps 8B align

__global__ __launch_bounds__(512)
void social_encoder_fused(const int*   __restrict__ nodes,
                          const int*   __restrict__ neigh_ids,
                          const int*   __restrict__ seg_ids,
                          const float* __restrict__ features,
                          const float* __restrict__ w1,     // (64, 128) row-major
                          const float* __restrict__ b1,     // (64,)
                          float*       __restrict__ out,    // (B, 64)
                          int n_edges)
{
    __shared__ float comb[TILE_M * LDS_STRIDE];   // 16 x 128 combined tile (padded)

    const int lane = threadIdx.x & 31;
    const int wave = threadIdx.x >> 5;            // 16 waves per block
    const int m0   = blockIdx.x * TILE_M;

    // ---------------- Phase 1: one wave aggregates one output row ----------------
    {
        const int b = m0 + wave;

        // self features: 64 floats, 2 per lane, fully coalesced 256B row load
        const int node = nodes[b];
        v2f selfv = *(const v2f*)(features + (size_t)node * EMBED + 2 * lane);

        // seg_ids is sorted: binary-search edge range [start, end) for row b
        int lo = 0, hi = n_edges;
        while (lo < hi) { int mid = (lo + hi) >> 1; if (seg_ids[mid] <  b)     lo = mid + 1; else hi = mid; }
        int lo2 = lo, hi2 = n_edges;
        while (lo2 < hi2) { int mid = (lo2 + hi2) >> 1; if (seg_ids[mid] < b + 1) lo2 = mid + 1; else hi2 = mid; }
        const int start = lo, end = lo2;
        const int count = end - start;

        v2f acc; acc.x = 0.0f; acc.y = 0.0f;
        for (int base = start; base < end; base += 32) {
            const int e  = base + lane;
            const int id = (e < end) ? neigh_ids[e] : 0;
            if (e < end) {
                // warm L2/L0 for the rows this wave is about to broadcast-read
                __builtin_prefetch(features + (size_t)id * EMBED,      0, 0);
                __builtin_prefetch(features + (size_t)id * EMBED + 32, 0, 0);
            }
            const int cnt = min(32, end - base);
            #pragma unroll 4
            for (int j = 0; j < cnt; ++j) {
                const int sid = __shfl(id, j, 32);
                v2f v = *(const v2f*)(features + (size_t)sid * EMBED + 2 * lane);
                acc.x += v.x;
                acc.y += v.y;
            }
        }
        const float inv = 1.0f / (float)max(count, 1);
        acc.x *= inv; acc.y *= inv;

        // combined[row] = [self(0:64) | neigh_mean(64:128)]
        *(v2f*)(&comb[wave * LDS_STRIDE + 2 * lane])         = selfv;
        *(v2f*)(&comb[wave * LDS_STRIDE + EMBED + 2 * lane]) = acc;
    }
    __syncthreads();

    // ---------------- Phase 2: waves 0..3 -> 16x64 output tile via WMMA f32 ------
    // out = relu(combined(16x128) @ w1^T(128x64) + b1); wave w owns N-tile n0=w*16
    if (wave < 4) {
        const int n0  = wave * TILE_M;
        const int nid = lane & 15;            // N index for B/C/D frags, M index for A
        const int kh  = (lane >> 4) * 2;      // A: lanes 0-15 hold K0,K1; 16-31 hold K2,K3

        v8f c = {};                           // 16x16 f32 accumulator (8 VGPRs)

        // B frag (4x16, K x N): lane n<16 -> B[k0+0..1][n]; lane>=16 -> B[k0+2..3][n-16]
        // B[k][n] = w1T[k][n0+n] = w1[(n0+n)*128 + k]  -> contiguous float2 per lane
        const float* wrow = w1   + (size_t)(n0 + nid) * KDIM + kh;
        const float* arow = comb + nid * LDS_STRIDE + kh;

        #pragma unroll
        for (int kk = 0; kk < KDIM / 4; ++kk) {
            const int k0 = kk * 4;
            v2f a = *(const v2f*)(arow + k0);   // ds_load_b64 from LDS tile
            v2f bf = *(const v2f*)(wrow + k0);  // global_load_b64, L2-hot (w1 = 32KB)
            c = __builtin_amdgcn_wmma_f32_16x16x4_f32(
                    /*neg_a=*/false, a, /*neg_b=*/false, bf,
                    /*c_mod=*/(short)0, c, /*reuse_a=*/false, /*reuse_b=*/false);
        }

        // D layout: VGPR r -> M = r (lanes 0-15) / M = r+8 (lanes 16-31), N = nid
        const float bias  = b1[n0 + nid];
        const int   mbase = m0 + (lane >> 4) * 8;
        #pragma unroll
        for (int r = 0; r < 8; ++r) {
            float v = c[r] + bias;
            v = v > 0.0f ? v : 0.0f;            // relu
            out[(size_t)(mbase + r) * EMBED + n0 + nid] = v;
        }
    }
}

extern "C" void kernel_launch(void* const* d_in, const int* in_sizes, int n_in,
                              void* d_out, int out_size, void* d_ws, size_t ws_size,
                              hipStream_t stream) {
    const int*   nodes     = (const int*)  d_in[0];   // (16384,)
    const int*   neigh_ids = (const int*)  d_in[1];   // (524288,)
    const int*   seg_ids   = (const int*)  d_in[2];   // (524288,) sorted
    const float* features  = (const float*)d_in[3];   // (1M, 64)
    const float* w1        = (const float*)d_in[4];   // (64, 128)
    const float* b1        = (const float*)d_in[5];   // (64,)
    float*       out       = (float*)d_out;           // (16384, 64)

    const int n_edges = in_sizes[1];
    const int batch   = in_sizes[0];
    const int blocks  = batch / TILE_M;               // 1024

    social_encoder_fused<<<blocks, 512, 0, stream>>>(
        nodes, neigh_ids, seg_ids, features, w1, b1, out, n_edges);
}